// QuantumLayer_65481071403616
// MI455X (gfx1250) — compile-verified
//
#include <hip/hip_runtime.h>

typedef float v2f __attribute__((ext_vector_type(2)));
typedef float v8f __attribute__((ext_vector_type(8)));

#define NQ 4
#define NL 3

// ---------------------------------------------------------------------------
// Phase 0 (tiny, 1 block): build the four real symmetric observables
//   A_i = Re(U^H D_i U), U = variational circuit unitary (16x16 complex).
// Threads 0..15 each simulate one column of U; then 256 threads compute the
// 4*256 entries of A_i into d_ws (A_i[a][b] at ws[i*256 + a*16 + b]).
// ---------------------------------------------------------------------------
__global__ void qprep_kernel(const float* __restrict__ params,
                             float* __restrict__ Aws) {
    __shared__ float Ure[16][16];
    __shared__ float Uim[16][16];
    const int tid = threadIdx.x;
    if (tid < 16) {
        float re[16], im[16];
        #pragma unroll
        for (int k = 0; k < 16; ++k) { re[k] = (k == tid) ? 1.f : 0.f; im[k] = 0.f; }
        #pragma unroll
        for (int l = 0; l < NL; ++l) {
            #pragma unroll
            for (int q = 0; q < NQ; ++q) {
                const float thy = params[(l * NQ + q) * 3 + 0];
                const float thz = params[(l * NQ + q) * 3 + 1];
                const int   m   = 8 >> q;           // bitmask of qubit q
                // RY(thy) on qubit q
                const float cy = cosf(0.5f * thy), sy = sinf(0.5f * thy);
                #pragma unroll
                for (int k = 0; k < 16; ++k) {
                    if ((k & m) == 0) {
                        const int k1 = k | m;
                        const float r0 = re[k], i0 = im[k], r1 = re[k1], i1 = im[k1];
                        re[k]  = cy * r0 - sy * r1;  im[k]  = cy * i0 - sy * i1;
                        re[k1] = sy * r0 + cy * r1;  im[k1] = sy * i0 + cy * i1;
                    }
                }
                // RZ(thz) on qubit q: bit0 *= (c - i s), bit1 *= (c + i s)
                const float cz = cosf(0.5f * thz), sz = sinf(0.5f * thz);
                #pragma unroll
                for (int k = 0; k < 16; ++k) {
                    const float r = re[k], i = im[k];
                    if ((k & m) == 0) { re[k] = r * cz + i * sz; im[k] = i * cz - r * sz; }
                    else              { re[k] = r * cz - i * sz; im[k] = i * cz + r * sz; }
                }
            }
            // CNOT chain 0->1, 1->2, 2->3
            #pragma unroll
            for (int c = 0; c < NQ - 1; ++c) {
                const int mc = 8 >> c, mt = 8 >> (c + 1);
                #pragma unroll
                for (int k = 0; k < 16; ++k) {
                    if ((k & mc) && !(k & mt)) {
                        const int k1 = k | mt;
                        float t;
                        t = re[k]; re[k] = re[k1]; re[k1] = t;
                        t = im[k]; im[k] = im[k1]; im[k1] = t;
                    }
                }
            }
        }
        #pragma unroll
        for (int k = 0; k < 16; ++k) { Ure[k][tid] = re[k]; Uim[k][tid] = im[k]; }
    }
    __syncthreads();
    const int a = tid >> 4, b = tid & 15;
    #pragma unroll
    for (int i = 0; i < NQ; ++i) {
        float acc = 0.f;
        #pragma unroll
        for (int k = 0; k < 16; ++k) {
            const float sgn = ((k >> (3 - i)) & 1) ? -1.f : 1.f;
            acc += sgn * (Ure[k][a] * Ure[k][b] + Uim[k][a] * Uim[k][b]);
        }
        Aws[i * 256 + a * 16 + b] = acc;
    }
}

// ---------------------------------------------------------------------------
// Phase 1: one wave32 processes tiles of 16 batch rows.
//   S (16 rows x 16 amplitudes, real) built per-lane from sin/cos of x.
//   D_i = A_i @ S^T  via 4 chained V_WMMA_F32_16X16X4_F32 per observable.
//   z_i[n] = sum_m S[n][m] * D_i[m][n]  -> 8-term register dot + 1 shfl_xor(16).
// ---------------------------------------------------------------------------
__global__ void __launch_bounds__(256)
qmain_kernel(const float* __restrict__ x, const float* __restrict__ Aws,
             float* __restrict__ out, int ntiles, int tpw) {
    const int lane = threadIdx.x & 31;
    const int h    = lane >> 4;      // lane half: selects K-offset +2 / b2 bit
    const int l15  = lane & 15;      // batch row within tile / matrix row
    const int wid  = (blockIdx.x * blockDim.x + threadIdx.x) >> 5;

    // Hoist constant A_i operands in WMMA A-layout:
    // chunk j, lane<16: {A_i[l15][4j], A_i[l15][4j+1]}; lane>=16: +2 offset.
    v2f amat[4][4];
    #pragma unroll
    for (int i = 0; i < 4; ++i) {
        #pragma unroll
        for (int j = 0; j < 4; ++j) {
            const float* p = Aws + i * 256 + l15 * 16 + 4 * j + 2 * h;
            amat[i][j] = v2f{p[0], p[1]};
        }
    }

    for (int t = 0; t < tpw; ++t) {
        const int tile = wid * tpw + t;          // uniform across the wave
        if (tile >= ntiles) break;
        const int base = tile * 16;

        // Each lane-pair (L, L+16) owns batch row l15: load its 4 angles.
        const float4 xv = ((const float4*)x)[base + l15];
        const float c0 = __cosf(0.5f * xv.x), s0 = __sinf(0.5f * xv.x);
        const float c1 = __cosf(0.5f * xv.y), s1 = __sinf(0.5f * xv.y);
        const float c2 = __cosf(0.5f * xv.z), s2 = __sinf(0.5f * xv.z);
        const float c3 = __cosf(0.5f * xv.w), s3 = __sinf(0.5f * xv.w);

        // S[r][k] = f0(b0) f1(b1) f2(b2) f3(b3), k = 8b0+4b1+2b2+b3.
        // This lane holds k = 4j + 2h + t2  (j = b0b1, t2 = b3, b2 = h).
        const float f2 = h ? s2 : c2;
        const float g0 = c0 * c1 * f2, g1 = c0 * s1 * f2;
        const float g2 = s0 * c1 * f2, g3 = s0 * s1 * f2;
        float sv[4][2];
        sv[0][0] = g0 * c3; sv[0][1] = g0 * s3;
        sv[1][0] = g1 * c3; sv[1][1] = g1 * s3;
        sv[2][0] = g2 * c3; sv[2][1] = g2 * s3;
        sv[3][0] = g3 * c3; sv[3][1] = g3 * s3;

        // Cross-half exchange so each lane holds S[l15][m] for m = 8h..8h+7.
        const float r00 = __shfl_xor(h ? sv[0][0] : sv[2][0], 16, 32);
        const float r01 = __shfl_xor(h ? sv[0][1] : sv[2][1], 16, 32);
        const float r10 = __shfl_xor(h ? sv[1][0] : sv[3][0], 16, 32);
        const float r11 = __shfl_xor(h ? sv[1][1] : sv[3][1], 16, 32);
        float w[8];
        w[0] = h ? r00      : sv[0][0];
        w[1] = h ? r01      : sv[0][1];
        w[2] = h ? sv[2][0] : r00;
        w[3] = h ? sv[2][1] : r01;
        w[4] = h ? r10      : sv[1][0];
        w[5] = h ? r11      : sv[1][1];
        w[6] = h ? sv[3][0] : r10;
        w[7] = h ? sv[3][1] : r11;

        // B-operand (S^T, K x N = amplitude x batch) chunk j is exactly sv[j].
        v2f bmat[4];
        #pragma unroll
        for (int j = 0; j < 4; ++j) bmat[j] = v2f{sv[j][0], sv[j][1]};

        float z[4];
        #pragma unroll
        for (int i = 0; i < 4; ++i) {
            v8f acc = {};
            #pragma unroll
            for (int j = 0; j < 4; ++j) {
                acc = __builtin_amdgcn_wmma_f32_16x16x4_f32(
                    /*neg_a=*/false, amat[i][j],
                    /*neg_b=*/false, bmat[j],
                    /*c_mod=*/(short)0, acc,
                    /*reuse_a=*/false, /*reuse_b=*/false);
            }
            // acc[v] = D_i[v + 8h][l15]; dot with this lane's S row slice.
            float p = 0.f;
            #pragma unroll
            for (int v = 0; v < 8; ++v) p += w[v] * acc[v];
            z[i] = p + __shfl_xor(p, 16, 32);    // add the other half's slice
        }

        if (h == 0) {
            ((float4*)out)[base + l15] = make_float4(z[0], z[1], z[2], z[3]);
        }
    }
}

// ---------------------------------------------------------------------------
extern "C" void kernel_launch(void* const* d_in, const int* in_sizes, int n_in,
                              void* d_out, int out_size, void* d_ws, size_t ws_size,
                              hipStream_t stream) {
    const float* x      = (const float*)d_in[0];   // (B, 4) float32
    const float* params = (const float*)d_in[1];   // (3, 4, 3) float32
    float*       out    = (float*)d_out;           // (B, 4) float32
    float*       Aws    = (float*)d_ws;            // 4 * 256 floats

    const int n      = in_sizes[0] / 4;            // batch size B
    const int ntiles = (n + 15) / 16;

    hipLaunchKernelGGL(qprep_kernel, dim3(1), dim3(256), 0, stream, params, Aws);

    const int TPW     = 4;                          // tiles per wave
    const int nwaves  = (ntiles + TPW - 1) / TPW;
    const int threads = 256;                        // 8 waves / block
    const int blocks  = (nwaves * 32 + threads - 1) / threads;
    hipLaunchKernelGGL(qmain_kernel, dim3(blocks), dim3(threads), 0, stream,
                       x, Aws, out, ntiles, TPW);
}